// MSAColumnAttention_22479858827762
// MI455X (gfx1250) — compile-verified
//
#include <hip/hip_runtime.h>
#include <hip/hip_bf16.h>

// Problem constants (from reference)
#define NSEQ 64      // S
#define NRES 128     // I
#define CDIM 64      // C
#define PDIM 32      // D
#define NHEAD 8      // H
#define NROWS (NSEQ * NRES)        // 8192 rows of x
#define QKV_M (3 * NHEAD * PDIM)   // 768
#define G_M   (NHEAD * PDIM)       // 256

typedef __attribute__((ext_vector_type(2))) float v2f;
typedef __attribute__((ext_vector_type(8))) float v8f;
typedef __attribute__((ext_vector_type(4))) unsigned int u32x4;
typedef __attribute__((ext_vector_type(8))) int i32x8;
typedef __attribute__((ext_vector_type(4))) int i32x4;

// ---------------------------------------------------------------------------
// Kernel 1: LayerNorm over last dim (C=64). One wave32 per row; each lane
// owns elements {lane, lane+32}; butterfly shuffle reduction for mean/var.
// ---------------------------------------------------------------------------
__global__ void ln_kernel(const float* __restrict__ x,
                          const float* __restrict__ sc,
                          const float* __restrict__ bi,
                          float* __restrict__ y) {
    const int row  = blockIdx.x;        // 0..8191
    const int lane = threadIdx.x;       // 0..31 (wave32)
    const float* xr = x + (size_t)row * CDIM;

    float a = xr[lane];
    float b = xr[lane + 32];
    float s  = a + b;
    float ss = a * a + b * b;
    #pragma unroll
    for (int off = 16; off > 0; off >>= 1) {
        s  += __shfl_xor(s,  off, 32);
        ss += __shfl_xor(ss, off, 32);
    }
    const float mu  = s * (1.0f / CDIM);
    const float var = ss * (1.0f / CDIM) - mu * mu;
    const float rinv = rsqrtf(var + 1e-5f);

    float* yr = y + (size_t)row * CDIM;
    yr[lane]      = (a - mu) * rinv * sc[lane]      + bi[lane];
    yr[lane + 32] = (b - mu) * rinv * sc[lane + 32] + bi[lane + 32];
}

// ---------------------------------------------------------------------------
// Kernel 2: fp32 GEMM  C[N,M] = A[N,K] @ B[K,M] (+ bias)
//
// One block (4 waves) per 16-row strip of A. The strip [16,K] is DMA'd into
// LDS once by the Tensor Data Mover (tensor_load_to_lds, TENSORcnt), with
// 1 dword of padding every 64 dwords so 16-lane fragment reads are LDS
// bank-conflict free (row stride K + K/64). Each wave then sweeps column
// tiles tn = wave, wave+4, ..., doing V_WMMA_F32_16X16X4_F32 with two
// interleaved accumulators to break the RAW chain.
//
// Fragment layouts (CDNA5 ISA 7.12.2, 32-bit data):
//   A 16x4 : lanes 0-15 rows m=lane, VGPR0=K0 VGPR1=K1; lanes 16-31 K2/K3.
//   B 4x16 : row-striped across lanes within a VGPR (mirrors C/D layout).
//   C 16x16: VGPR j: lanes0-15 row m=j, lanes16-31 row m=j+8.
// ---------------------------------------------------------------------------
__global__ void __launch_bounds__(128)
gemm_wmma_f32(const float* __restrict__ A,   // [N,K]
              const float* __restrict__ B,   // [K,M]
              const float* __restrict__ bias,// [M] or null
              float* __restrict__ C,         // [N,M]
              int N, int K, int M) {
    __shared__ float lds_a[16 * 260];        // max: 16 rows * (256 + 4) floats

    const int wave = threadIdx.x >> 5;       // 0..3
    const int lane = threadIdx.x & 31;
    const int row0 = blockIdx.x << 4;        // 16-row strip
    const int Kp   = K + (K >> 6);           // padded LDS row stride (floats)

    if (wave == 0) {
        // --- Tensor DMA descriptor (D#), 2-D tensor: 16 x K fp32 tile ---
        const unsigned long long ga =
            (unsigned long long)(const void*)(A + (size_t)row0 * K);
        const unsigned int lds_off =
            (unsigned int)(unsigned long long)(const void*)lds_a; // low 32 bits = LDS byte offset

        u32x4 g0;
        g0[0] = 1u;                                   // count=1, user descriptor
        g0[1] = lds_off;                              // lds_addr (bytes)
        g0[2] = (unsigned int)(ga & 0xFFFFFFFFull);   // global_addr[31:0]
        g0[3] = (unsigned int)((ga >> 32) & 0x1FFFFFFull) | (2u << 30); // addr[56:32] | type=2

        i32x8 g1;
        g1[0] = (2 << 16)        // data_size = 4 bytes
              | (1 << 20)        // pad_enable
              | (5 << 22)        // pad_interval: 64 dwords
              | (0 << 25);       // pad_amount: 1 dword
        g1[1] = (K & 0xFFFF) << 16;                   // tensor_dim0[15:0]
        g1[2] = ((K >> 16) & 0xFFFF) | (16 << 16);    // tensor_dim0[31:16] | tensor_dim1 lo
        g1[3] = (K << 16);                            // tensor_dim1 hi | tile_dim0 = K
        g1[4] = 16;                                   // tile_dim1 = 16, tile_dim2 = 0
        g1[5] = K;                                    // tensor_dim0_stride[31:0]
        g1[6] = 0;                                    // dim0_stride hi | dim1_stride lo
        g1[7] = 0;                                    // tensor_dim1_stride hi

        i32x4 gz4 = {};                               // groups 2/3 unused (2-D)
        i32x8 gz8 = {};                               // extra group (clang-23 6-arg form)
        __builtin_amdgcn_tensor_load_to_lds(g0, g1, gz4, gz4, gz8, 0);
        __builtin_amdgcn_s_wait_tensorcnt(0);
    }
    __syncthreads();

    const int half = lane >> 4;              // 0 or 1
    const int l15  = lane & 15;
    const int tilesM = M >> 4;

    for (int tn = wave; tn < tilesM; tn += 4) {       // wave-uniform loop
        const int col0 = tn << 4;
        v8f acc0 = {};
        v8f acc1 = {};
        const float* Bp0 = B + col0 + l15;

        for (int k8 = 0; k8 < K; k8 += 8) {
            // --- even 4-slice ---
            {
                const int ka = k8 + 2 * half;
                const int ia = l15 * Kp + ka + (ka >> 6);
                v2f a; a.x = lds_a[ia]; a.y = lds_a[ia + 1];
                const float* Bp = Bp0 + (size_t)ka * M;
                v2f b; b.x = Bp[0]; b.y = Bp[M];
                __builtin_prefetch(Bp + 8 * (size_t)M, 0, 1); // global_prefetch_b8
                acc0 = __builtin_amdgcn_wmma_f32_16x16x4_f32(
                           false, a, false, b, (short)0, acc0, false, false);
            }
            // --- odd 4-slice (independent accumulator: breaks RAW chain) ---
            {
                const int ka = k8 + 4 + 2 * half;
                const int ia = l15 * Kp + ka + (ka >> 6);
                v2f a; a.x = lds_a[ia]; a.y = lds_a[ia + 1];
                const float* Bp = Bp0 + (size_t)ka * M;
                v2f b; b.x = Bp[0]; b.y = Bp[M];
                acc1 = __builtin_amdgcn_wmma_f32_16x16x4_f32(
                           false, a, false, b, (short)0, acc1, false, false);
            }
        }

        #pragma unroll
        for (int j = 0; j < 8; ++j) {
            const int r = row0 + j + 8 * half;
            const int c = col0 + l15;
            float vout = acc0[j] + acc1[j];
            if (bias) vout += bias[c];
            C[(size_t)r * M + c] = vout;
        }
    }
}

// ---------------------------------------------------------------------------
// Kernel 3: per-feature column attention + gating.
// One wave per (i,h); lane = d. For each feature f = d*H + h the S=64
// q/k/v vectors live at stride I*768 in the qkv buffer. k,v kept in
// registers (128 VGPRs); max_t(q_s*k_t) via precomputed kmax/kmin.
// ---------------------------------------------------------------------------
__global__ void attn_kernel(const float* __restrict__ qkv,   // [8192, 768]
                            const float* __restrict__ glin,  // [8192, 256]
                            float* __restrict__ ctx) {       // [8192, 256]
    const int i = blockIdx.x;                 // 0..127
    const int h = threadIdx.x >> 5;           // 0..7
    const int d = threadIdx.x & 31;           // lane = d
    const int f = d * NHEAD + h;              // feature index in 256-dim

    const float invs = 0.17677669529663687f;  // 1/sqrt(32)
    const size_t rs = (size_t)NRES * QKV_M;   // row stride between s values

    const float* qp = qkv + (size_t)i * QKV_M + 0 * G_M + f;
    const float* kp = qkv + (size_t)i * QKV_M + 1 * G_M + f;
    const float* vp = qkv + (size_t)i * QKV_M + 2 * G_M + f;

    float kr[NSEQ], vr[NSEQ];
    #pragma unroll
    for (int t = 0; t < NSEQ; ++t) {
        kr[t] = kp[(size_t)t * rs] * invs;    // fold 1/sqrt(D) into k
        vr[t] = vp[(size_t)t * rs];
    }
    float kmax = kr[0], kmin = kr[0];
    #pragma unroll
    for (int t = 1; t < NSEQ; ++t) {
        kmax = fmaxf(kmax, kr[t]);
        kmin = fminf(kmin, kr[t]);
    }

    for (int s = 0; s < NSEQ; ++s) {
        const float qs = qp[(size_t)s * rs];
        const float m  = (qs >= 0.0f) ? qs * kmax : qs * kmin;  // max_t score
        float den = 0.0f, num = 0.0f;
        #pragma unroll
        for (int t = 0; t < NSEQ; ++t) {
            const float e = __expf(fmaf(qs, kr[t], -m));        // v_exp_f32
            den += e;
            num = fmaf(e, vr[t], num);
        }
        const size_t orow = (size_t)(s * NRES + i);
        const float gl = glin[orow * G_M + f];
        const float gate = 1.0f / (1.0f + __expf(-gl));
        ctx[orow * G_M + f] = gate * num / den;
    }
}

// ---------------------------------------------------------------------------
// Launcher: LN -> QKV gemm -> gate gemm -> attention -> out gemm (+bias)
// Workspace: x_ln 2MB | qkv 24MB | glin 8MB | ctx 8MB  (42MB total)
// ---------------------------------------------------------------------------
extern "C" void kernel_launch(void* const* d_in, const int* in_sizes, int n_in,
                              void* d_out, int out_size, void* d_ws, size_t ws_size,
                              hipStream_t stream) {
    const float* msa   = (const float*)d_in[0];
    const float* ln_s  = (const float*)d_in[1];
    const float* ln_b  = (const float*)d_in[2];
    const float* W_qkv = (const float*)d_in[3];
    const float* W_g   = (const float*)d_in[4];
    const float* W_out = (const float*)d_in[5];
    const float* b_out = (const float*)d_in[6];
    float* out = (float*)d_out;

    float* x_ln = (float*)d_ws;
    float* qkvb = x_ln + (size_t)NROWS * CDIM;
    float* glin = qkvb + (size_t)NROWS * QKV_M;
    float* ctx  = glin + (size_t)NROWS * G_M;

    // 1) LayerNorm: one wave per row
    ln_kernel<<<NROWS, 32, 0, stream>>>(msa, ln_s, ln_b, x_ln);

    // 2) QKV projection: [8192,64] @ [64,768]  (TDM-staged A strip per block)
    gemm_wmma_f32<<<NROWS / 16, 128, 0, stream>>>(
        x_ln, W_qkv, nullptr, qkvb, NROWS, CDIM, QKV_M);

    // 3) Gate projection: [8192,64] @ [64,256]
    gemm_wmma_f32<<<NROWS / 16, 128, 0, stream>>>(
        x_ln, W_g, nullptr, glin, NROWS, CDIM, G_M);

    // 4) Per-feature attention + sigmoid gate: 128 blocks x (8 waves = 8 heads)
    attn_kernel<<<NRES, NHEAD * 32, 0, stream>>>(qkvb, glin, ctx);

    // 5) Output projection: [8192,256] @ [256,64] + b_out
    gemm_wmma_f32<<<NROWS / 16, 128, 0, stream>>>(
        ctx, W_out, b_out, out, NROWS, G_M, CDIM);
}